// _PointnetSAModuleFSBase_7954279432426
// MI455X (gfx1250) — compile-verified
//
#include <hip/hip_runtime.h>

// ---- problem config (matches reference) ----
#define NB    2
#define NPTS  16384
#define CINF  64
#define NP    2048
#define NS    32
#define RAD2  0.64f        // 0.8^2
#define C1    64
#define C2    128
#define C3    256
#define BNRS  0.99999500003749968f   // 1/sqrt(1+1e-5), eval-mode BN scale

typedef float v2f __attribute__((ext_vector_type(2)));
typedef float v8f __attribute__((ext_vector_type(8)));

__device__ __forceinline__ v8f wmma4(v2f a, v2f b, v8f c) {
  // V_WMMA_F32_16X16X4_F32 : D(16x16 f32) = A(16x4) * B(4x16) + C
  return __builtin_amdgcn_wmma_f32_16x16x4_f32(false, a, false, b, (short)0, c,
                                               false, false);
}

__device__ __forceinline__ float reluf(float x) { return x > 0.f ? x : 0.f; }

// ---------------- Stage 1: furthest point sampling (all-LDS resident) --------
// 1024 threads, dynamic LDS: mind[NPTS] + xs/ys/zs[NPTS] + 32-entry reduce
// Fused "update min-dist + argmax" single pass per round; wave32 shfl butterfly
// reductions; only the 3-float result per round touches global memory.
__global__ void fps_kernel(const float* __restrict__ xyz, float* __restrict__ newxyz) {
  extern __shared__ float lds[];
  float* md = lds;                     // NPTS
  float* xs = lds + NPTS;              // NPTS
  float* ys = lds + 2 * NPTS;          // NPTS
  float* zs = lds + 3 * NPTS;          // NPTS
  float* rv = lds + 4 * NPTS;          // 32
  int*   ri = (int*)(lds + 4 * NPTS + 32);  // 32
  const int b = blockIdx.x;
  const int tid = threadIdx.x;         // 1024 threads = 32 waves
  const int lane = tid & 31, wave = tid >> 5;
  const float* X = xyz + (size_t)b * NPTS * 3;
  float* nx = newxyz + (size_t)b * NP * 3;

  for (int i = tid; i < NPTS; i += 1024) {
    xs[i] = X[3 * i]; ys[i] = X[3 * i + 1]; zs[i] = X[3 * i + 2];
    md[i] = 3.4e38f;
  }
  __syncthreads();
  float cx = xs[0], cy = ys[0], cz = zs[0];
  if (tid == 0) { nx[0] = cx; nx[1] = cy; nx[2] = cz; }

  for (int t = 1; t < NP; ++t) {
    float bv = -1.f; int bi = NPTS;
    for (int i = tid; i < NPTS; i += 1024) {
      float dx = xs[i] - cx, dy = ys[i] - cy, dz = zs[i] - cz;
      float d = dx * dx + dy * dy + dz * dz;
      float m = md[i];
      m = d < m ? d : m;
      md[i] = m;
      if (m > bv || (m == bv && i < bi)) { bv = m; bi = i; }
    }
    // intra-wave argmax (value desc, index asc tiebreak to match jnp.argmax)
    for (int off = 16; off > 0; off >>= 1) {
      float ov = __shfl_xor(bv, off, 32);
      int   oi = __shfl_xor(bi, off, 32);
      if (ov > bv || (ov == bv && oi < bi)) { bv = ov; bi = oi; }
    }
    if (lane == 0) { rv[wave] = bv; ri[wave] = bi; }
    __syncthreads();
    if (wave == 0) {
      bv = rv[lane]; bi = ri[lane];
      for (int off = 16; off > 0; off >>= 1) {
        float ov = __shfl_xor(bv, off, 32);
        int   oi = __shfl_xor(bi, off, 32);
        if (ov > bv || (ov == bv && oi < bi)) { bv = ov; bi = oi; }
      }
      if (lane == 0) ri[0] = bi;
    }
    __syncthreads();
    const int nxt = ri[0];
    cx = xs[nxt]; cy = ys[nxt]; cz = zs[nxt];
    if (tid == 0) { nx[3 * t] = cx; nx[3 * t + 1] = cy; nx[3 * t + 2] = cz; }
    __syncthreads();   // everyone consumed ri[0] before next round overwrites
  }
}
#define FPS_LDS_BYTES ((4 * NPTS + 64) * 4)

// ---------------- Stage 2: ball query (one wave32 per center) ----------------
__global__ void ballq_kernel(const float* __restrict__ xyz, const float* __restrict__ newxyz,
                             int* __restrict__ nn, int* __restrict__ cnt) {
  const int wave = threadIdx.x >> 5;
  const int lane = threadIdx.x & 31;
  const int ctr = blockIdx.x * 4 + wave;   // 0 .. NB*NP-1
  const int b = ctr / NP;
  const float* X = xyz + (size_t)b * NPTS * 3;
  const float cx = newxyz[ctr * 3], cy = newxyz[ctr * 3 + 1], cz = newxyz[ctr * 3 + 2];
  int* out = nn + (size_t)ctr * NS;
  int found = 0, total = 0, first = 0;
  for (int base = 0; base < NPTS; base += 32) {
    const int i = base + lane;
    float dx = X[3 * i] - cx, dy = X[3 * i + 1] - cy, dz = X[3 * i + 2] - cz;
    float d2 = dx * dx + dy * dy + dz * dz;
    unsigned m = __builtin_amdgcn_ballot_w32(d2 < RAD2);
    total += __popc(m);
    while (m && found < NS) {
      int pos = __ffs(m) - 1;
      int idx = base + pos;
      if (found == 0) first = idx;
      if (lane == 0) out[found] = idx;
      ++found;
      m &= m - 1;
    }
  }
  if (lane == 0) {
    for (int t2 = found; t2 < NS; ++t2) out[t2] = first;  // pad with first hit (0 if none)
    cnt[ctr] = total;
  }
}

// ------- Stage 3: gather + 3-layer WMMA MLP + masked max-pool (1 center/WG) -------
__global__ void group_mlp_kernel(const float* __restrict__ xyz, const float* __restrict__ feat,
                                 const float* __restrict__ newxyz,
                                 const int* __restrict__ nn, const int* __restrict__ cnt,
                                 const float* __restrict__ W1, const float* __restrict__ g1, const float* __restrict__ b1,
                                 const float* __restrict__ W2, const float* __restrict__ g2, const float* __restrict__ b2,
                                 const float* __restrict__ W3, const float* __restrict__ g3, const float* __restrict__ b3,
                                 float* __restrict__ xw) {
  __shared__ float Xin[NS][68];   // 67 cols + zero pad
  __shared__ float Y1[NS][C1];
  __shared__ float Y2[NS][C2];
  const int blk = blockIdx.x;     // b*NP + j
  const int b = blk / NP;
  const int j = blk - b * NP;
  const int tid = threadIdx.x;
  const int wave = tid >> 5, lane = tid & 31;
  const int col = lane & 15, half = lane >> 4;
  const int* nnp = nn + (size_t)blk * NS;
  const float cx = newxyz[blk * 3], cy = newxyz[blk * 3 + 1], cz = newxyz[blk * 3 + 2];

  for (int t = tid; t < NS * 68; t += 128) {
    const int r = t / 68, c = t - r * 68;
    float v = 0.f;
    if (c < 67) {
      const int pi = nnp[r];
      if (c < 3) {
        float p = xyz[((size_t)b * NPTS + pi) * 3 + c];
        v = p - (c == 0 ? cx : (c == 1 ? cy : cz));
      } else {
        v = feat[((size_t)b * CINF + (c - 3)) * NPTS + pi];
      }
    }
    Xin[r][c] = v;
  }
  __syncthreads();

  // layer 1: 32x67 @ 67x64
  for (int tile = wave; tile < 8; tile += 4) {
    const int mt = tile >> 2, nt = tile & 3;
    const int o = nt * 16 + col;
    v8f acc = {};
    for (int k0 = 0; k0 < 68; k0 += 4) {
      const int ka = k0 + 2 * half;
      v2f a; a.x = Xin[mt * 16 + col][ka]; a.y = Xin[mt * 16 + col][ka + 1];
      v2f w; w.x = (ka < 67) ? W1[o * 67 + ka] : 0.f;
             w.y = (ka + 1 < 67) ? W1[o * 67 + ka + 1] : 0.f;
      acc = wmma4(a, w, acc);
    }
    const float sc = g1[o] * BNRS, bi = b1[o];
    for (int r = 0; r < 8; ++r)
      Y1[mt * 16 + r + 8 * half][o] = reluf(acc[r] * sc + bi);
  }
  __syncthreads();

  // layer 2: 32x64 @ 64x128
  for (int tile = wave; tile < 16; tile += 4) {
    const int mt = tile >> 3, nt = tile & 7;
    const int o = nt * 16 + col;
    v8f acc = {};
    for (int k0 = 0; k0 < C1; k0 += 4) {
      const int ka = k0 + 2 * half;
      v2f a; a.x = Y1[mt * 16 + col][ka]; a.y = Y1[mt * 16 + col][ka + 1];
      v2f w; w.x = W2[o * C1 + ka]; w.y = W2[o * C1 + ka + 1];
      acc = wmma4(a, w, acc);
    }
    const float sc = g2[o] * BNRS, bi = b2[o];
    for (int r = 0; r < 8; ++r)
      Y2[mt * 16 + r + 8 * half][o] = reluf(acc[r] * sc + bi);
  }
  __syncthreads();

  // layer 3: 32x128 @ 128x256 + masked max-pool over the 32 samples
  const int nball = cnt[blk];
  for (int nt = wave; nt < 16; nt += 4) {
    const int o = nt * 16 + col;
    v8f acc0 = {}, acc1 = {};
    for (int k0 = 0; k0 < C2; k0 += 4) {
      const int ka = k0 + 2 * half;
      v2f a0; a0.x = Y2[col][ka];      a0.y = Y2[col][ka + 1];
      v2f a1; a1.x = Y2[16 + col][ka]; a1.y = Y2[16 + col][ka + 1];
      v2f w;  w.x = W3[o * C2 + ka];   w.y = W3[o * C2 + ka + 1];
      acc0 = wmma4(a0, w, acc0);
      acc1 = wmma4(a1, w, acc1);
    }
    const float sc = g3[o] * BNRS, bi = b3[o];
    float m = 0.f;   // relu outputs are >= 0
    for (int r = 0; r < 8; ++r) {
      m = fmaxf(m, reluf(acc0[r] * sc + bi));
      m = fmaxf(m, reluf(acc1[r] * sc + bi));
    }
    m = fmaxf(m, __shfl_xor(m, 16, 32));   // combine row halves
    if (nball == 0) m = 0.f;               // mask empty balls
    if (lane < 16)
      xw[((size_t)b * C3 + o) * NP + j] = m;
  }
}

// ---------------- Stage 4: squeeze max + SE excitation ----------------
__global__ void smax_kernel(const float* __restrict__ xw, float* __restrict__ sw) {
  const int row = blockIdx.x;          // b*C3 + c
  const float* xr = xw + (size_t)row * NP;
  __shared__ float red[256];
  float m = 0.f;                        // x >= 0
  for (int i = threadIdx.x; i < NP; i += 256) m = fmaxf(m, xr[i]);
  red[threadIdx.x] = m;
  __syncthreads();
  for (int s = 128; s > 0; s >>= 1) {
    if (threadIdx.x < s) red[threadIdx.x] = fmaxf(red[threadIdx.x], red[threadIdx.x + s]);
    __syncthreads();
  }
  if (threadIdx.x == 0) sw[row] = red[0];
}

__global__ void excite_kernel(const float* __restrict__ sw, const float* __restrict__ We1,
                              const float* __restrict__ We2, float* __restrict__ ew) {
  const int b = blockIdx.x;
  const int tid = threadIdx.x;
  __shared__ float sl[C3];
  __shared__ float h32[32];
  sl[tid] = sw[b * C3 + tid];
  __syncthreads();
  if (tid < 32) {
    float a = 0.f;
    for (int i = 0; i < C3; ++i) a += We1[tid * C3 + i] * sl[i];
    h32[tid] = reluf(a);
  }
  __syncthreads();
  float a = 0.f;
  for (int i = 0; i < 32; ++i) a += We2[tid * 32 + i] * h32[i];
  ew[b * C3 + tid] = 1.f / (1.f + expf(-a));
}

// ---------------- Stage 5: q/k projections (WMMA, K=2048) ----------------
__global__ void qk_kernel(const float* __restrict__ xw,
                          const float* __restrict__ Wq, const float* __restrict__ gq, const float* __restrict__ bq,
                          const float* __restrict__ Wk, const float* __restrict__ gk, const float* __restrict__ bk,
                          float* __restrict__ qw, float* __restrict__ kw) {
  const int id = blockIdx.x;
  const int tile = id & 255;
  const int which = (id >> 8) & 1;
  const int b = id >> 9;
  const int mt = tile >> 4, nt = tile & 15;
  const int lane = threadIdx.x & 31;
  const int col = lane & 15, half = lane >> 4;
  const float* W = which ? Wk : Wq;
  const float* g = which ? gk : gq;
  const float* bb = which ? bk : bq;
  float* out = which ? kw : qw;
  const float* xb = xw + (size_t)b * C3 * NP;
  const int m0 = mt * 16 + col, n0 = nt * 16 + col;
  v8f acc = {};
  for (int k0 = 0; k0 < NP; k0 += 4) {
    const int ka = k0 + 2 * half;
    v2f a;  a.x = W[(size_t)m0 * NP + ka];  a.y = W[(size_t)m0 * NP + ka + 1];
    v2f bv; bv.x = xb[(size_t)n0 * NP + ka]; bv.y = xb[(size_t)n0 * NP + ka + 1];
    acc = wmma4(a, bv, acc);
  }
  for (int r = 0; r < 8; ++r) {
    const int o = mt * 16 + r + 8 * half;
    float v = reluf(acc[r] * (g[o] * BNRS) + bb[o]);
    out[(size_t)b * C3 * C3 + (size_t)o * C3 + (nt * 16 + col)] = v;
  }
}

// ---------------- Stage 6: sim = k^T q (WMMA) ----------------
__global__ void sim_kernel(const float* __restrict__ qw, const float* __restrict__ kw,
                           float* __restrict__ simw) {
  const int id = blockIdx.x;
  const int tile = id & 255;
  const int b = id >> 8;
  const int mt = tile >> 4, nt = tile & 15;
  const int lane = threadIdx.x & 31;
  const int col = lane & 15, half = lane >> 4;
  const float* qb = qw + (size_t)b * C3 * C3;
  const float* kb = kw + (size_t)b * C3 * C3;
  const int m0 = mt * 16 + col, n0 = nt * 16 + col;
  v8f acc = {};
  for (int k0 = 0; k0 < C3; k0 += 4) {
    const int ka = k0 + 2 * half;
    v2f a;  a.x = kb[(size_t)ka * C3 + m0];  a.y = kb[(size_t)(ka + 1) * C3 + m0];
    v2f bv; bv.x = qb[(size_t)ka * C3 + n0]; bv.y = qb[(size_t)(ka + 1) * C3 + n0];
    acc = wmma4(a, bv, acc);
  }
  for (int r = 0; r < 8; ++r)
    simw[(size_t)b * C3 * C3 + (size_t)(mt * 16 + r + 8 * half) * C3 + n0] = acc[r];
}

// ---------------- Stage 7: aff = softmax(max - sim) == softmax(-sim) ----------------
__global__ void softmax_kernel(float* __restrict__ simw) {
  const int row = blockIdx.x;         // b*C3 + c
  float* sr = simw + (size_t)row * C3;
  const int tid = threadIdx.x;
  __shared__ float red[C3];
  const float v = sr[tid];
  red[tid] = v;
  __syncthreads();
  for (int s = 128; s > 0; s >>= 1) {
    if (tid < s) red[tid] = fminf(red[tid], red[tid + s]);
    __syncthreads();
  }
  const float mn = red[0];
  __syncthreads();
  const float w = expf(mn - v);
  red[tid] = w;
  __syncthreads();
  for (int s = 128; s > 0; s >>= 1) {
    if (tid < s) red[tid] += red[tid + s];
    __syncthreads();
  }
  sr[tid] = w / red[0];
}

// ---------------- Stage 8: out = alpha * aff @ (x*e) + x (WMMA) ----------------
__global__ void out_kernel(const float* __restrict__ affw, const float* __restrict__ xw,
                           const float* __restrict__ ew, const float* __restrict__ alpha,
                           float* __restrict__ outp) {
  const int id = blockIdx.x;
  const int tile = id & 2047;
  const int b = id >> 11;
  const int mt = tile >> 7, nt = tile & 127;
  const int lane = threadIdx.x & 31;
  const int col = lane & 15, half = lane >> 4;
  const float* ab = affw + (size_t)b * C3 * C3;
  const float* xb = xw + (size_t)b * C3 * NP;
  const float* eb = ew + b * C3;
  const int m0 = mt * 16 + col, n0 = nt * 16 + col;
  v8f acc = {};
  for (int k0 = 0; k0 < C3; k0 += 4) {
    const int ka = k0 + 2 * half;
    v2f a;  a.x = ab[(size_t)m0 * C3 + ka]; a.y = ab[(size_t)m0 * C3 + ka + 1];
    v2f bv; bv.x = xb[(size_t)ka * NP + n0] * eb[ka];
            bv.y = xb[(size_t)(ka + 1) * NP + n0] * eb[ka + 1];
    acc = wmma4(a, bv, acc);
  }
  const float al = alpha[0];
  for (int r = 0; r < 8; ++r) {
    const int o = mt * 16 + r + 8 * half;
    const size_t idx = (size_t)b * C3 * NP + (size_t)o * NP + n0;
    outp[idx] = al * acc[r] + xb[(size_t)o * NP + n0];
  }
}

extern "C" void kernel_launch(void* const* d_in, const int* in_sizes, int n_in,
                              void* d_out, int out_size, void* d_ws, size_t ws_size,
                              hipStream_t stream) {
  (void)in_sizes; (void)n_in; (void)out_size; (void)ws_size;
  const float* xyz  = (const float*)d_in[0];
  const float* feat = (const float*)d_in[1];
  const float* W1 = (const float*)d_in[2];
  const float* g1 = (const float*)d_in[3];
  const float* b1 = (const float*)d_in[4];
  const float* W2 = (const float*)d_in[5];
  const float* g2 = (const float*)d_in[6];
  const float* b2 = (const float*)d_in[7];
  const float* W3 = (const float*)d_in[8];
  const float* g3 = (const float*)d_in[9];
  const float* b3 = (const float*)d_in[10];
  const float* Wq = (const float*)d_in[11];
  const float* gq = (const float*)d_in[12];
  const float* bq = (const float*)d_in[13];
  const float* Wk = (const float*)d_in[14];
  const float* gk = (const float*)d_in[15];
  const float* bk = (const float*)d_in[16];
  const float* We1 = (const float*)d_in[17];
  const float* We2 = (const float*)d_in[18];
  const float* alpha = (const float*)d_in[19];

  float* newxyz = (float*)d_out;                          // (NB,NP,3)
  float* outp   = (float*)d_out + (size_t)NB * NP * 3;    // (NB,C3,NP)

  char* ws = (char*)d_ws;
  int*   nn   = (int*)ws;    ws += (size_t)NB * NP * NS * 4;
  int*   cnt  = (int*)ws;    ws += (size_t)NB * NP * 4;
  float* xw   = (float*)ws;  ws += (size_t)NB * C3 * NP * 4;
  float* qw   = (float*)ws;  ws += (size_t)NB * C3 * C3 * 4;
  float* kw   = (float*)ws;  ws += (size_t)NB * C3 * C3 * 4;
  float* simw = (float*)ws;  ws += (size_t)NB * C3 * C3 * 4;
  float* sw   = (float*)ws;  ws += (size_t)NB * C3 * 4;
  float* ew   = (float*)ws;  ws += (size_t)NB * C3 * 4;

  fps_kernel<<<NB, 1024, FPS_LDS_BYTES, stream>>>(xyz, newxyz);
  ballq_kernel<<<NB * NP / 4, 128, 0, stream>>>(xyz, newxyz, nn, cnt);
  group_mlp_kernel<<<NB * NP, 128, 0, stream>>>(xyz, feat, newxyz, nn, cnt,
                                                W1, g1, b1, W2, g2, b2, W3, g3, b3, xw);
  smax_kernel<<<NB * C3, 256, 0, stream>>>(xw, sw);
  excite_kernel<<<NB, C3, 0, stream>>>(sw, We1, We2, ew);
  qk_kernel<<<NB * 2 * 256, 32, 0, stream>>>(xw, Wq, gq, bq, Wk, gk, bk, qw, kw);
  sim_kernel<<<NB * 256, 32, 0, stream>>>(qw, kw, simw);
  softmax_kernel<<<NB * C3, C3, 0, stream>>>(simw);
  out_kernel<<<NB * 2048, 32, 0, stream>>>(simw, xw, ew, alpha, outp);
}